// EfficientAttention_32220844654841
// MI455X (gfx1250) — compile-verified
//
#include <hip/hip_runtime.h>
#include <hip/hip_bf16.h>

#define BB   64
#define DIMC 128
#define NPT  4000
#define NH   4
#define DH   32
#define CL   50
#define KNN  9
#define SCAL 0.17677669529663687f   // 32^-0.5

typedef __attribute__((ext_vector_type(2))) float v2f;
typedef __attribute__((ext_vector_type(8))) float v8f;

__device__ __forceinline__ v8f wmma4(v2f a, v2f b, v8f c) {
  // D = A(16x4,f32) * B(4x16,f32) + C(16x16,f32)
  return __builtin_amdgcn_wmma_f32_16x16x4_f32(false, a, false, b, (short)0, c, false, false);
}
__device__ __forceinline__ v8f v8zero() {
  v8f v;
#pragma unroll
  for (int i = 0; i < 8; ++i) v[i] = 0.f;
  return v;
}

// ---------- 1: per-(b,c) instance-norm stats over n ----------
__global__ void k_row_stats(const float* __restrict__ x, float* __restrict__ mu,
                            float* __restrict__ rs, float* __restrict__ ratio) {
  const int row = blockIdx.x;                       // b*DIMC + c
  const float* p = x + (size_t)row * NPT;
  float s = 0.f, ss = 0.f;
  for (int i = threadIdx.x; i < NPT; i += 256) { float v = p[i]; s += v; ss += v * v; }
  __shared__ float shs[256], shq[256];
  shs[threadIdx.x] = s; shq[threadIdx.x] = ss;
  __syncthreads();
  for (int o = 128; o > 0; o >>= 1) {
    if (threadIdx.x < o) { shs[threadIdx.x] += shs[threadIdx.x + o]; shq[threadIdx.x] += shq[threadIdx.x + o]; }
    __syncthreads();
  }
  if (threadIdx.x == 0) {
    float m = shs[0] * (1.f / NPT);
    float var = fmaxf(shq[0] * (1.f / NPT) - m * m, 0.f);
    mu[row] = m;
    rs[row] = rsqrtf(var + 1e-3f);
    ratio[row] = var / (var + 1e-3f);                // = E_n[xin^2]
  }
}

// ---------- 2: fold IN+BN into per-(b,c) affine coeffs for both groups ----------
__global__ void k_coeffs(const float* __restrict__ mu, const float* __restrict__ rs,
                         const float* __restrict__ ratio,
                         const float* __restrict__ g1g, const float* __restrict__ g1b,
                         const float* __restrict__ g2g, const float* __restrict__ g2b,
                         float* __restrict__ ca1, float* __restrict__ cd1,
                         float* __restrict__ ca2, float* __restrict__ cd2) {
  int c = threadIdx.x;
  if (c >= DIMC) return;
  float s = 0.f;
  for (int b = 0; b < BB; ++b) s += ratio[b * DIMC + c];
  float rbn = rsqrtf(s * (1.f / BB) + 1e-5f);        // BN mean over (b,n) is exactly 0
  float s1 = g1g[c] * rbn, s2 = g2g[c] * rbn;
  for (int b = 0; b < BB; ++b) {
    int i = b * DIMC + c;
    float r = rs[i], m = mu[i];
    float A1 = s1 * r; ca1[i] = A1; cd1[i] = g1b[c] - A1 * m;
    float A2 = s2 * r; ca2[i] = A2; cd2[i] = g2b[c] - A2 * m;
  }
}

// ---------- 3: fused embed GEMM (both groups), WMMA f32 16x16x4 ----------
// e_g[b,cl,n] = sum_c wg[cl,c]*relu(a_bc*x+d_bc) + bg[cl]
// Weight tiles padded to 64 rows of zeros in LDS -> no exec-masked LDS loads in hot loop.
__global__ __launch_bounds__(256) void k_embed(
    const float* __restrict__ x,
    const float* __restrict__ ca1, const float* __restrict__ cd1,
    const float* __restrict__ ca2, const float* __restrict__ cd2,
    const float* __restrict__ wg1, const float* __restrict__ bg1,
    const float* __restrict__ wg2, const float* __restrict__ bg2,
    float* __restrict__ e1, float* __restrict__ e2) {
  const int b = blockIdx.y;
  __shared__ float lw1[64][DIMC], lw2[64][DIMC];     // rows 50..63 = 0
  __shared__ float lc[4][DIMC];
  __shared__ float lb[2][CL];
  const int tid = threadIdx.x;
  for (int i = tid; i < 64 * DIMC; i += 256) {
    const int r = i >> 7;                            // row (i / DIMC)
    const bool v = (r < CL);
    ((float*)lw1)[i] = v ? wg1[i] : 0.f;
    ((float*)lw2)[i] = v ? wg2[i] : 0.f;
  }
  for (int i = tid; i < DIMC; i += 256) {
    lc[0][i] = ca1[b * DIMC + i]; lc[1][i] = cd1[b * DIMC + i];
    lc[2][i] = ca2[b * DIMC + i]; lc[3][i] = cd2[b * DIMC + i];
  }
  for (int i = tid; i < CL; i += 256) { lb[0][i] = bg1[i]; lb[1][i] = bg2[i]; }
  __syncthreads();
  const int wave = tid >> 5, lane = tid & 31, q = lane & 15, kh = lane >> 4;
  const int base = blockIdx.x * 128 + wave * 16;     // 16 columns per wave
  if (base >= NPT) return;                           // wave-uniform
  const int ncol = base + q;
  const float* xc = x + (size_t)b * DIMC * NPT + ncol;
  v8f acc1[4], acc2[4];
#pragma unroll
  for (int i = 0; i < 4; ++i) { acc1[i] = v8zero(); acc2[i] = v8zero(); }
  for (int t = 0; t < 32; ++t) {                     // K=128 in steps of 4
    const int kb = 4 * t + 2 * kh;
    float x0 = xc[(size_t)kb * NPT];
    float x1 = xc[(size_t)(kb + 1) * NPT];
    if (t < 31) {                                    // gfx1250 global_prefetch_b8
      __builtin_prefetch(xc + (size_t)(kb + 4) * NPT, 0, 1);
    }
    v2f b1v, b2v;
    b1v.x = fmaxf(lc[0][kb] * x0 + lc[1][kb], 0.f);
    b1v.y = fmaxf(lc[0][kb + 1] * x1 + lc[1][kb + 1], 0.f);
    b2v.x = fmaxf(lc[2][kb] * x0 + lc[3][kb], 0.f);
    b2v.y = fmaxf(lc[2][kb + 1] * x1 + lc[3][kb + 1], 0.f);
#pragma unroll
    for (int mt = 0; mt < 4; ++mt) {                 // 64 cluster rows (zero-padded)
      const int m = mt * 16 + q;
      v2f a1v, a2v;
      a1v.x = lw1[m][kb]; a1v.y = lw1[m][kb + 1];
      a2v.x = lw2[m][kb]; a2v.y = lw2[m][kb + 1];
      acc1[mt] = wmma4(a1v, b1v, acc1[mt]);
      acc2[mt] = wmma4(a2v, b2v, acc2[mt]);
    }
  }
#pragma unroll
  for (int mt = 0; mt < 4; ++mt)
#pragma unroll
    for (int r = 0; r < 8; ++r) {
      const int clr = mt * 16 + 8 * kh + r;
      if (clr < CL) {
        e1[((size_t)b * CL + clr) * NPT + ncol] = acc1[mt][r] + lb[0][clr];
        e2[((size_t)b * CL + clr) * NPT + ncol] = acc2[mt][r] + lb[1][clr];
      }
    }
}

// ---------- 4: softmax over n (rows of 4000), in place ----------
__global__ void k_softmax_n(float* __restrict__ e) {
  float* p = e + (size_t)blockIdx.x * NPT;
  __shared__ float sh[256];
  float m = -3.4e38f;
  for (int i = threadIdx.x; i < NPT; i += 256) m = fmaxf(m, p[i]);
  sh[threadIdx.x] = m; __syncthreads();
  for (int o = 128; o > 0; o >>= 1) { if (threadIdx.x < o) sh[threadIdx.x] = fmaxf(sh[threadIdx.x], sh[threadIdx.x + o]); __syncthreads(); }
  m = sh[0]; __syncthreads();
  float s = 0.f;
  for (int i = threadIdx.x; i < NPT; i += 256) s += __expf(p[i] - m);
  sh[threadIdx.x] = s; __syncthreads();
  for (int o = 128; o > 0; o >>= 1) { if (threadIdx.x < o) sh[threadIdx.x] += sh[threadIdx.x + o]; __syncthreads(); }
  float r = 1.f / sh[0];
  for (int i = threadIdx.x; i < NPT; i += 256) p[i] = __expf(p[i] - m) * r;
}

// ---------- 5: cluster_x[b,d,cl] = sum_n x[b,d,n]*S1[b,cl,n]  (WMMA, LDS-staged) ----------
__global__ __launch_bounds__(256) void k_clusterx(const float* __restrict__ x,
                                                  const float* __restrict__ s1,
                                                  float* __restrict__ cx) {
  const int b = blockIdx.x;
  __shared__ float lx[DIMC][64];
  __shared__ float ls[64][64];
  const int tid = threadIdx.x;
  const int wave = tid >> 5, lane = tid & 31, q = lane & 15, kh = lane >> 4;
  v8f acc[4];
#pragma unroll
  for (int i = 0; i < 4; ++i) acc[i] = v8zero();
  const int nch = (NPT + 63) / 64;
  for (int c = 0; c < nch; ++c) {
    const int nb = c * 64;
    __syncthreads();
    for (int i = tid; i < DIMC * 64; i += 256) {
      int rr = i >> 6, jj = i & 63, n = nb + jj;
      lx[rr][jj] = (n < NPT) ? x[((size_t)b * DIMC + rr) * NPT + n] : 0.f;
    }
    for (int i = tid; i < 64 * 64; i += 256) {
      int rr = i >> 6, jj = i & 63, n = nb + jj;
      ls[rr][jj] = (rr < CL && n < NPT) ? s1[((size_t)b * CL + rr) * NPT + n] : 0.f;
    }
    __syncthreads();
#pragma unroll 4
    for (int kk = 0; kk < 16; ++kk) {
      const int kb = 4 * kk + 2 * kh;
      v2f aa; aa.x = lx[wave * 16 + q][kb]; aa.y = lx[wave * 16 + q][kb + 1];
#pragma unroll
      for (int nt = 0; nt < 4; ++nt) {
        v2f bb; bb.x = ls[nt * 16 + q][kb]; bb.y = ls[nt * 16 + q][kb + 1];
        acc[nt] = wmma4(aa, bb, acc[nt]);
      }
    }
  }
#pragma unroll
  for (int nt = 0; nt < 4; ++nt)
#pragma unroll
    for (int r = 0; r < 8; ++r) {
      int d = wave * 16 + 8 * kh + r;
      int cl = nt * 16 + q;
      if (cl < CL) cx[((size_t)b * DIMC + d) * CL + cl] = acc[nt][r];
    }
}

// ---------- 6: q + qkv1 projections, stacked: qall[b,h,0:128,cl] ----------
__global__ void k_qall(const float* __restrict__ cx,
                       const float* __restrict__ qkvw, const float* __restrict__ qkvb,
                       const float* __restrict__ qkv1w, const float* __restrict__ qkv1b,
                       float* __restrict__ qall) {
  size_t gid = (size_t)blockIdx.x * 256 + threadIdx.x;
  if (gid >= (size_t)BB * NH * 128 * CL) return;
  int cl = gid % CL; size_t t = gid / CL;
  int o4 = t % 128; t /= 128;
  int h = t % NH; int b = (int)(t / NH);
  const float* w; float bias;
  if (o4 < DH) { w = qkvw + ((size_t)h * DH + o4) * DIMC; bias = qkvb[h * DH + o4]; }
  else { w = qkv1w + ((size_t)h * 3 * DH + (o4 - DH)) * DIMC; bias = qkv1b[h * 3 * DH + (o4 - DH)]; }
  const float* cp = cx + (size_t)b * DIMC * CL + cl;
  float acc = bias;
  for (int c = 0; c < DIMC; ++c) acc += w[c] * cp[(size_t)c * CL];
  qall[gid] = acc;
}

// ---------- 7: knn(9) graph + conv1 (pre-BN) ----------
__global__ void k_local(const float* __restrict__ qall, const float* __restrict__ gc1w,
                        const float* __restrict__ gc1b, float* __restrict__ h1pre) {
  const int bh = blockIdx.x, h = bh % NH;
  __shared__ float xt[CL][DH];
  __shared__ float sq[CL];
  __shared__ float w1[DH][2 * DH][3];
  __shared__ float spd[CL][CL];
  const int tid = threadIdx.x;
  for (int i = tid; i < CL * DH; i += 64) { int n = i / DH, d = i % DH; xt[n][d] = qall[((size_t)bh * 128 + d) * CL + n]; }
  for (int i = tid; i < DH * 2 * DH * 3; i += 64) ((float*)w1)[i] = gc1w[(size_t)h * DH * 2 * DH * 3 + i];
  __syncthreads();
  if (tid < CL) { float s = 0.f; for (int d = 0; d < DH; ++d) s += xt[tid][d] * xt[tid][d]; sq[tid] = s; }
  __syncthreads();
  if (tid < CL) {
    const int n = tid;
    for (int m = 0; m < CL; ++m) {
      float in = 0.f;
      for (int d = 0; d < DH; ++d) in += xt[n][d] * xt[m][d];
      spd[n][m] = 2.f * in - sq[n] - sq[m];
    }
    int idx[KNN]; unsigned long long used = 0ull;
    for (int k = 0; k < KNN; ++k) {                  // top-9, ties -> lower index (jax)
      float best = -3.4e38f; int bi = 0;
      for (int m = 0; m < CL; ++m)
        if (((used >> m) & 1ull) == 0 && spd[n][m] > best) { best = spd[n][m]; bi = m; }
      used |= (1ull << bi); idx[k] = bi;
    }
    for (int j = 0; j < 3; ++j)
      for (int o = 0; o < DH; ++o) {
        float acc = gc1b[h * DH + o];
        for (int t = 0; t < 3; ++t) {
          int m = idx[j * 3 + t];
          for (int i2 = 0; i2 < DH; ++i2) {
            float ci = xt[n][i2];
            acc += ci * w1[o][i2][t];
            acc += (ci - xt[m][i2]) * w1[o][i2 + DH][t];
          }
        }
        h1pre[(((size_t)bh * CL + n) * 3 + j) * DH + o] = acc;
      }
  }
}

// ---------- 8/10: BN(train) stats over (b, n[, j]) per (h,o); J=3 or 1 ----------
__global__ void k_bn_stats(const float* __restrict__ src, const float* __restrict__ gamma,
                           const float* __restrict__ beta, float* __restrict__ scale,
                           float* __restrict__ shift, int J) {
  const int ho = blockIdx.x, h = ho / DH, o = ho % DH;
  const int cnt = BB * CL * J;
  float s = 0.f, ss = 0.f;
  for (int idx = threadIdx.x; idx < cnt; idx += 256) {
    int j = idx % J, n = (idx / J) % CL, b = idx / (J * CL);
    float v = src[(((size_t)(b * NH + h) * CL + n) * J + j) * DH + o];
    s += v; ss += v * v;
  }
  __shared__ float shs[256], shq[256];
  shs[threadIdx.x] = s; shq[threadIdx.x] = ss;
  __syncthreads();
  for (int w = 128; w > 0; w >>= 1) {
    if (threadIdx.x < w) { shs[threadIdx.x] += shs[threadIdx.x + w]; shq[threadIdx.x] += shq[threadIdx.x + w]; }
    __syncthreads();
  }
  if (threadIdx.x == 0) {
    float m = shs[0] / cnt;
    float var = fmaxf(shq[0] / cnt - m * m, 0.f);
    float sc = gamma[ho] * rsqrtf(var + 1e-5f);
    scale[ho] = sc; shift[ho] = beta[ho] - sc * m;
  }
}

// ---------- 9: conv2 (pre-BN) ----------
__global__ void k_conv2(const float* __restrict__ h1pre, const float* __restrict__ sc,
                        const float* __restrict__ sh, const float* __restrict__ gc2w,
                        const float* __restrict__ gc2b, float* __restrict__ h2pre) {
  size_t gid = (size_t)blockIdx.x * 256 + threadIdx.x;
  if (gid >= (size_t)BB * NH * CL * DH) return;
  int o = gid % DH; size_t t = gid / DH;
  int n = t % CL; t /= CL;
  int h = t % NH; int b = (int)(t / NH);
  int bh = b * NH + h;
  float acc = gc2b[h * DH + o];
  for (int tt = 0; tt < 3; ++tt)
    for (int i = 0; i < DH; ++i) {
      float v = h1pre[(((size_t)bh * CL + n) * 3 + tt) * DH + i];
      v = fmaxf(sc[h * DH + i] * v + sh[h * DH + i], 0.f);
      acc += v * gc2w[(((size_t)h * DH + o) * DH + i) * 3 + tt];
    }
  h2pre[gid] = acc;
}

// ---------- 11: tiny attention over 50 clusters + assemble cat[b,256,cl] ----------
__global__ void k_attn_cat(const float* __restrict__ qall, const float* __restrict__ h2pre,
                           const float* __restrict__ sc2, const float* __restrict__ sh2,
                           float* __restrict__ cat) {
  const int bh = blockIdx.x, b = bh / NH, h = bh % NH;
  __shared__ float lq[DH][CL], lk[DH][CL], lv[DH][CL];
  __shared__ float sl[CL][CL];
  const int tid = threadIdx.x;
  for (int i = tid; i < DH * CL; i += 64) {
    int d = i / CL, n = i % CL;
    lq[d][n] = qall[((size_t)bh * 128 + 32 + d) * CL + n];
    lk[d][n] = qall[((size_t)bh * 128 + 64 + d) * CL + n];
    lv[d][n] = qall[((size_t)bh * 128 + 96 + d) * CL + n];
  }
  __syncthreads();
  if (tid < CL) {
    const int n = tid;
    float mx = -3.4e38f;
    for (int m = 0; m < CL; ++m) {
      float lg = 0.f;
      for (int d = 0; d < DH; ++d) lg += lq[d][n] * lk[d][m];
      lg *= SCAL;
      sl[n][m] = lg; mx = fmaxf(mx, lg);
    }
    float s = 0.f;
    for (int m = 0; m < CL; ++m) s += __expf(sl[n][m] - mx);
    float r = 1.f / s;
    for (int m = 0; m < CL; ++m) sl[n][m] = __expf(sl[n][m] - mx) * r;
    for (int d = 0; d < DH; ++d) {
      float acc = 0.f;
      for (int m = 0; m < CL; ++m) acc += sl[n][m] * lv[d][m];
      cat[((size_t)b * 2 * DIMC + h * 64 + 32 + d) * CL + n] = acc;
    }
    for (int o = 0; o < DH; ++o) {
      float v = h2pre[((size_t)bh * CL + n) * DH + o];
      v = fmaxf(sc2[h * DH + o] * v + sh2[h * DH + o], 0.f);
      cat[((size_t)b * 2 * DIMC + h * 64 + o) * CL + n] = v;
    }
  }
}

// ---------- 12: res = pres_w @ cat + pres_b ----------
__global__ void k_res(const float* __restrict__ cat, const float* __restrict__ presw,
                      const float* __restrict__ presb, float* __restrict__ res) {
  size_t gid = (size_t)blockIdx.x * 256 + threadIdx.x;
  if (gid >= (size_t)BB * DIMC * CL) return;
  int cl = gid % CL; size_t t = gid / CL;
  int o = t % DIMC; int b = (int)(t / DIMC);
  float acc = presb[o];
  const float* wp = presw + (size_t)o * 2 * DIMC;
  const float* cp = cat + (size_t)b * 2 * DIMC * CL + cl;
  for (int i = 0; i < 2 * DIMC; ++i) acc += wp[i] * cp[(size_t)i * CL];
  res[gid] = acc;
}

// ---------- 13: M = proj_w @ res (bias folded into final kernel) ----------
__global__ void k_mproj(const float* __restrict__ res, const float* __restrict__ projw,
                        float* __restrict__ M) {
  size_t gid = (size_t)blockIdx.x * 256 + threadIdx.x;
  if (gid >= (size_t)BB * DIMC * CL) return;
  int cl = gid % CL; size_t t = gid / CL;
  int o = t % DIMC; int b = (int)(t / DIMC);
  float acc = 0.f;
  for (int d = 0; d < DIMC; ++d) acc += projw[(size_t)o * DIMC + d] * res[((size_t)b * DIMC + d) * CL + cl];
  M[gid] = acc;
}

// ---------- 14: out[b,o,n] = sum_cl M[b,o,cl]*softmax_cl(embed2)[cl,n] + proj_b (WMMA) ----------
__global__ __launch_bounds__(128) void k_out(const float* __restrict__ e2,
                                             const float* __restrict__ M,
                                             const float* __restrict__ projb,
                                             float* __restrict__ out) {
  const int b = blockIdx.y;
  const int base = blockIdx.x * 64;
  __shared__ float lm[DIMC][64];
  __shared__ float ls2[64][64];
  __shared__ float lpb[DIMC];
  const int tid = threadIdx.x;
  for (int i = tid; i < DIMC * 64; i += 128) {
    int o = i >> 6, cl = i & 63;
    lm[o][cl] = (cl < CL) ? M[((size_t)b * DIMC + o) * CL + cl] : 0.f;
  }
  for (int i = tid; i < DIMC; i += 128) lpb[i] = projb[i];
  if (tid < 64) {
    int col = base + tid;
    int cc = col < NPT ? col : (NPT - 1);
    float mx = -3.4e38f;
    for (int cl = 0; cl < CL; ++cl) {
      float v = e2[((size_t)b * CL + cl) * NPT + cc];
      ls2[cl][tid] = v; mx = fmaxf(mx, v);
    }
    float s = 0.f;
    for (int cl = 0; cl < CL; ++cl) s += __expf(ls2[cl][tid] - mx);
    float r = 1.f / s;
    for (int cl = 0; cl < CL; ++cl) ls2[cl][tid] = __expf(ls2[cl][tid] - mx) * r;
    for (int cl = CL; cl < 64; ++cl) ls2[cl][tid] = 0.f;
  }
  __syncthreads();
  const int wave = tid >> 5, lane = tid & 31, q = lane & 15, kh = lane >> 4;
  const int coll = wave * 16 + q;
  const int n = base + coll;
  v8f acc[8];
#pragma unroll
  for (int i = 0; i < 8; ++i) acc[i] = v8zero();
#pragma unroll 4
  for (int kk = 0; kk < 16; ++kk) {
    const int kb = 4 * kk + 2 * kh;
    v2f bb; bb.x = ls2[kb][coll]; bb.y = ls2[kb + 1][coll];
#pragma unroll
    for (int mt = 0; mt < 8; ++mt) {
      v2f aa; aa.x = lm[mt * 16 + q][kb]; aa.y = lm[mt * 16 + q][kb + 1];
      acc[mt] = wmma4(aa, bb, acc[mt]);
    }
  }
  if (n < NPT) {
#pragma unroll
    for (int mt = 0; mt < 8; ++mt)
#pragma unroll
      for (int r = 0; r < 8; ++r) {
        int o = mt * 16 + 8 * kh + r;
        out[((size_t)b * DIMC + o) * NPT + n] = acc[mt][r] + lpb[o];
      }
  }
}

extern "C" void kernel_launch(void* const* d_in, const int* in_sizes, int n_in,
                              void* d_out, int out_size, void* d_ws, size_t ws_size,
                              hipStream_t stream) {
  (void)in_sizes; (void)n_in; (void)out_size; (void)ws_size;
  const float* x     = (const float*)d_in[0];
  const float* wg1   = (const float*)d_in[1];
  const float* bg1   = (const float*)d_in[2];
  const float* g1g   = (const float*)d_in[3];
  const float* g1b   = (const float*)d_in[4];
  const float* wg2   = (const float*)d_in[5];
  const float* bg2   = (const float*)d_in[6];
  const float* g2g   = (const float*)d_in[7];
  const float* g2b   = (const float*)d_in[8];
  const float* qkvw  = (const float*)d_in[9];
  const float* qkvb  = (const float*)d_in[10];
  const float* qkv1w = (const float*)d_in[11];
  const float* qkv1b = (const float*)d_in[12];
  const float* gc1w  = (const float*)d_in[13];
  const float* gc1b  = (const float*)d_in[14];
  const float* gc1g  = (const float*)d_in[15];
  const float* gc1be = (const float*)d_in[16];
  const float* gc2w  = (const float*)d_in[17];
  const float* gc2b  = (const float*)d_in[18];
  const float* gc2g  = (const float*)d_in[19];
  const float* gc2be = (const float*)d_in[20];
  const float* presw = (const float*)d_in[21];
  const float* presb = (const float*)d_in[22];
  const float* projw = (const float*)d_in[23];
  const float* projb = (const float*)d_in[24];

  float* w = (float*)d_ws;
  const size_t RC = (size_t)BB * DIMC;               // 8192
  float* mu    = w;                 w += RC;
  float* rs    = w;                 w += RC;
  float* ratio = w;                 w += RC;
  float* ca1   = w;                 w += RC;
  float* cd1   = w;                 w += RC;
  float* ca2   = w;                 w += RC;
  float* cd2   = w;                 w += RC;
  float* e1    = w;                 w += (size_t)BB * CL * NPT;   // 12.8M (S1 in place)
  float* e2    = w;                 w += (size_t)BB * CL * NPT;   // 12.8M
  float* cx    = w;                 w += (size_t)BB * DIMC * CL;
  float* qall  = w;                 w += (size_t)BB * NH * 128 * CL;
  float* h1    = w;                 w += (size_t)BB * NH * CL * 3 * DH;
  float* h2    = w;                 w += (size_t)BB * NH * CL * DH;
  float* s1s   = w;                 w += NH * DH;
  float* t1s   = w;                 w += NH * DH;
  float* s2s   = w;                 w += NH * DH;
  float* t2s   = w;                 w += NH * DH;
  float* cat   = w;                 w += (size_t)BB * 2 * DIMC * CL;
  float* res   = w;                 w += (size_t)BB * DIMC * CL;
  float* Mb    = w;                 w += (size_t)BB * DIMC * CL;

  k_row_stats<<<BB * DIMC, 256, 0, stream>>>(x, mu, rs, ratio);
  k_coeffs<<<1, 128, 0, stream>>>(mu, rs, ratio, g1g, g1b, g2g, g2b, ca1, cd1, ca2, cd2);
  k_embed<<<dim3(32, BB), 256, 0, stream>>>(x, ca1, cd1, ca2, cd2, wg1, bg1, wg2, bg2, e1, e2);
  k_softmax_n<<<BB * CL, 256, 0, stream>>>(e1);
  k_clusterx<<<BB, 256, 0, stream>>>(x, e1, cx);
  {
    size_t tot = (size_t)BB * NH * 128 * CL;
    k_qall<<<(unsigned)((tot + 255) / 256), 256, 0, stream>>>(cx, qkvw, qkvb, qkv1w, qkv1b, qall);
  }
  k_local<<<BB * NH, 64, 0, stream>>>(qall, gc1w, gc1b, h1);
  k_bn_stats<<<NH * DH, 256, 0, stream>>>(h1, gc1g, gc1be, s1s, t1s, 3);
  {
    size_t tot = (size_t)BB * NH * CL * DH;
    k_conv2<<<(unsigned)((tot + 255) / 256), 256, 0, stream>>>(h1, s1s, t1s, gc2w, gc2b, h2);
  }
  k_bn_stats<<<NH * DH, 256, 0, stream>>>(h2, gc2g, gc2be, s2s, t2s, 1);
  k_attn_cat<<<BB * NH, 64, 0, stream>>>(qall, h2, s2s, t2s, cat);
  {
    size_t tot = (size_t)BB * DIMC * CL;
    k_res<<<(unsigned)((tot + 255) / 256), 256, 0, stream>>>(cat, presw, presb, res);
    k_mproj<<<(unsigned)((tot + 255) / 256), 256, 0, stream>>>(res, projw, Mb);
  }
  k_out<<<dim3((NPT + 63) / 64, BB), 128, 0, stream>>>(e2, Mb, projb, (float*)d_out);
}